// SkipGramMultiContext_90254442758235
// MI455X (gfx1250) — compile-verified
//
#include <hip/hip_runtime.h>
#include <hip/hip_bf16.h>

// ---------------------------------------------------------------------------
// SkipGram multi-context NEG loss on MI455X (gfx1250, wave32).
//
// One wave32 handles one sample:
//   - 15 dot products (10 ctx + 5 neg) of length D=128, computed in full f32
//     via V_WMMA_F32_16X16X4_F32 (A = target row broadcast into all 16 rows,
//     B column N = pair N's out_emb row). 32 accumulating WMMAs over K=128,
//     software-pipelined against the global_load_b64 gather stream.
//   - Because A's rows are identical, every D element in lane L equals
//     dot(target, pair_of_column(L)); lane L reads its own pair's score.
//   - Epilogue: masked -logsigmoid using hardware v_exp_f32/v_log_f32
//     (arg of log is in [1,2] -> well-conditioned), wave shuffle reduction,
//     per-block partials to d_ws, deterministic tree reduction -> 2 scalars.
// ---------------------------------------------------------------------------

#define DIMS   128
#define C_MAX  10
#define K_NEG  5
#define WAVES_PER_BLOCK 8   // 256 threads = 8 wave32

typedef __attribute__((ext_vector_type(2))) float v2f;
typedef __attribute__((ext_vector_type(8))) float v8f;

__device__ __forceinline__ float neg_log_sigmoid(float x) {
    // -log(sigmoid(x)) = max(-x, 0) + log(1 + exp(-|x|))
    // exp(-|x|) in (0,1] so the log argument is in [1,2]: v_log_f32 is
    // well-conditioned there; no need for the heavy log1pf expansion.
    float e = __expf(-__builtin_fabsf(x));
    return fmaxf(-x, 0.0f) + __logf(1.0f + e);
}

__global__ void __launch_bounds__(WAVES_PER_BLOCK * 32)
sg_wmma_kernel(const int* __restrict__ target_words,
               const int* __restrict__ context_words,
               const int* __restrict__ context_lens,
               const int* __restrict__ negative_words,
               const float* __restrict__ in_emb,
               const float* __restrict__ out_emb,
               float* __restrict__ partials,   // [gridDim.x][2]
               int B) {
    __shared__ float sPos[WAVES_PER_BLOCK];
    __shared__ float sNeg[WAVES_PER_BLOCK];

    const int tid    = threadIdx.x;
    const int wave   = tid >> 5;
    const int lane   = tid & 31;
    const int sample = blockIdx.x * WAVES_PER_BLOCK + wave;

    float pos_per = 0.0f, neg_per = 0.0f;

    if (sample < B) {
        const int N  = lane & 15;     // matrix column == pair index
        const int hi = lane >> 4;     // half-wave: covers K elements {2hi, 2hi+1}

        // Pair word for this lane's column (column 15 is a dummy).
        int w;
        if (N < C_MAX)            w = context_words[sample * C_MAX + N];
        else if (N < C_MAX+K_NEG) w = negative_words[sample * K_NEG + (N - C_MAX)];
        else                      w = 0;

        const float* __restrict__ trow = in_emb  + (size_t)target_words[sample] * DIMS;
        const float* __restrict__ prow = out_emb + (size_t)w * DIMS;
        const int e = 2 * hi;

        v8f acc = {};
        #pragma unroll 8
        for (int k = 0; k < DIMS; k += 4) {
            // A-frag: target chunk (same per lane-half -> all 16 rows identical)
            v2f a = *(const v2f*)(trow + k + e);
            // B-frag: this lane's pair chunk (8B aligned global_load_b64)
            v2f b = *(const v2f*)(prow + k + e);
            // D = A*B + C, full f32:  v_wmma_f32_16x16x4_f32
            acc = __builtin_amdgcn_wmma_f32_16x16x4_f32(
                false, a, false, b, (short)0, acc, false, false);
        }

        // Every row of D is dot(target, column N); take VGPR0 of the frag.
        const float score = acc[0];
        const int   clen  = context_lens[sample];

        float posc = 0.0f, negc = 0.0f;
        if (lane < 16) {   // lanes 16..31 are duplicate copies of columns 0..15
            if (N < C_MAX) {
                if (N < clen) posc = neg_log_sigmoid(score);
            } else if (N < C_MAX + K_NEG) {
                negc = neg_log_sigmoid(-score);
            }
        }
        // wave32 reduction
        #pragma unroll
        for (int off = 16; off > 0; off >>= 1) {
            posc += __shfl_down(posc, off, 32);
            negc += __shfl_down(negc, off, 32);
        }
        if (lane == 0) {
            if (clen > 0) {
                pos_per = posc / (float)clen;
                neg_per = negc * (1.0f / K_NEG);
            }
        }
    }

    if (lane == 0) { sPos[wave] = pos_per; sNeg[wave] = neg_per; }
    __syncthreads();

    if (tid == 0) {
        float p = 0.0f, n = 0.0f;
        #pragma unroll
        for (int i = 0; i < WAVES_PER_BLOCK; ++i) { p += sPos[i]; n += sNeg[i]; }
        partials[2 * blockIdx.x + 0] = p;
        partials[2 * blockIdx.x + 1] = n;
    }
}

__global__ void __launch_bounds__(256)
sg_reduce_kernel(const float* __restrict__ partials, float* __restrict__ out,
                 int nblocks, float invB) {
    __shared__ float sp[256];
    __shared__ float sn[256];
    float p = 0.0f, n = 0.0f;
    for (int i = threadIdx.x; i < nblocks; i += 256) {
        p += partials[2 * i + 0];
        n += partials[2 * i + 1];
    }
    sp[threadIdx.x] = p;
    sn[threadIdx.x] = n;
    __syncthreads();
    for (int s = 128; s > 0; s >>= 1) {
        if (threadIdx.x < s) {
            sp[threadIdx.x] += sp[threadIdx.x + s];
            sn[threadIdx.x] += sn[threadIdx.x + s];
        }
        __syncthreads();
    }
    if (threadIdx.x == 0) {
        out[0] = sp[0] * invB;   // positive_loss
        out[1] = sn[0] * invB;   // negative_loss
    }
}

extern "C" void kernel_launch(void* const* d_in, const int* in_sizes, int n_in,
                              void* d_out, int out_size, void* d_ws, size_t ws_size,
                              hipStream_t stream) {
    const int*   target_words   = (const int*)  d_in[0];
    const int*   context_words  = (const int*)  d_in[1];
    const int*   context_lens   = (const int*)  d_in[2];
    const int*   negative_words = (const int*)  d_in[3];
    const float* in_emb         = (const float*)d_in[4];
    const float* out_emb        = (const float*)d_in[5];
    float*       out            = (float*)d_out;
    float*       partials       = (float*)d_ws;

    const int B       = in_sizes[0];
    const int nblocks = (B + WAVES_PER_BLOCK - 1) / WAVES_PER_BLOCK;

    sg_wmma_kernel<<<nblocks, WAVES_PER_BLOCK * 32, 0, stream>>>(
        target_words, context_words, context_lens, negative_words,
        in_emb, out_emb, partials, B);

    sg_reduce_kernel<<<1, 256, 0, stream>>>(partials, out, nblocks, 1.0f / (float)B);
}